// VINeuralODE_54769422958763
// MI455X (gfx1250) — compile-verified
//
#include <hip/hip_runtime.h>
#include <math.h>

#define DD 128
#define HH 512
#define BB 128
#define NSTEPS 16
#define ROWS 16
#define NWG (BB / ROWS)   // 8 workgroups, fully independent
#define TPB 256           // 8 wave32 per workgroup
#define XP 130            // padded row stride for 128-wide LDS arrays
#define TP 520            // padded row stride for 512-wide LDS array

typedef __attribute__((ext_vector_type(2))) float v2f;
typedef __attribute__((ext_vector_type(8))) float v8f;

// Dopri5 tableau
__device__ __constant__ float cAt[5][5] = {
    {1.f / 5.f, 0.f, 0.f, 0.f, 0.f},
    {3.f / 40.f, 9.f / 40.f, 0.f, 0.f, 0.f},
    {44.f / 45.f, -56.f / 15.f, 32.f / 9.f, 0.f, 0.f},
    {19372.f / 6561.f, -25360.f / 2187.f, 64448.f / 6561.f, -212.f / 729.f, 0.f},
    {9017.f / 3168.f, -355.f / 33.f, 46732.f / 5247.f, 49.f / 176.f, -5103.f / 18656.f}};
__device__ __constant__ float cC[6] = {0.f, 1.f / 5.f, 3.f / 10.f, 4.f / 5.f, 8.f / 9.f, 1.f};
__device__ __constant__ float cBW[6] = {35.f / 384.f, 0.f, 500.f / 1113.f,
                                        125.f / 192.f, -2187.f / 6784.f, 11.f / 84.f};

static __device__ inline v8f wmma_f32(v2f a, v2f b, v8f c) {
  // V_WMMA_F32_16X16X4_F32 : D(16x16 f32) = A(16x4 f32) * B(4x16 f32) + C
  return __builtin_amdgcn_wmma_f32_16x16x4_f32(false, a, false, b, (short)0, c,
                                               false, false);
}

extern "C" __global__ __launch_bounds__(TPB)
void VINeuralODE_54769422958763_kernel(const float* __restrict__ x0,
                                       const float* __restrict__ W1,
                                       const float* __restrict__ b1,
                                       const float* __restrict__ u1,
                                       const float* __restrict__ W2,
                                       const float* __restrict__ b2,
                                       float* __restrict__ out,
                                       volatile float* __restrict__ kxg) {
  __shared__ __align__(16) float s_xcur[ROWS * XP];   // current x, 16 rows
  __shared__ __align__(16) float s_xs[ROWS * XP];     // stage input x
  __shared__ __align__(16) float s_th[ROWS * TP];     // tanh activations 16x512
  __shared__ __align__(16) float s_kxi[ROWS * XP];    // current stage dxdt
  __shared__ __align__(16) float s_c[HH];             // trace weights c_k
  __shared__ float s_redtr[TPB];
  __shared__ float s_reddot[TPB];
  __shared__ float s_kld[6 * ROWS];
  __shared__ float s_kkl[6 * ROWS];
  __shared__ float s_ld[ROWS];
  __shared__ float s_kl[ROWS];

  const int tid = threadIdx.x;
  const int wg = blockIdx.x;          // 0..7 : which 16-row batch slice
  const int row0 = wg * ROWS;
  const int lane = tid & 31;
  const int wave = tid >> 5;          // 0..7
  const int mrow = lane & 15;
  const int hi = lane >> 4;
  const float dt = 1.0f / (float)NSTEPS;

  // ---- init: x state, trace weights c_k = sum_i W1[i,k]*W2[k,i] ----
  for (int e = tid; e < ROWS * DD; e += TPB)
    s_xcur[(e >> 7) * XP + (e & 127)] = x0[row0 * DD + e];
  for (int k = tid; k < HH; k += TPB) {
    float acc = 0.f;
    for (int i = 0; i < DD; ++i) acc += W1[i * HH + k] * W2[k * DD + i];
    s_c[k] = acc;
  }
  if (tid < ROWS) { s_ld[tid] = 0.f; s_kl[tid] = 0.f; }
  __syncthreads();

  for (int s = 0; s < NSTEPS; ++s) {
    const float t = (float)s * dt;
    for (int st = 0; st < 6; ++st) {
      // ---- stage input: xs = xcur + dt * sum_j A[st-1][j] * k_j (row-local) ----
      if (st == 0) {
        for (int e = tid; e < ROWS * DD; e += TPB) {
          int li = (e >> 7) * XP + (e & 127);
          s_xs[li] = s_xcur[li];
        }
      } else {
        for (int e = tid; e < ROWS * DD; e += TPB) {
          int li = (e >> 7) * XP + (e & 127);
          float acc = s_xcur[li];
          for (int j = 0; j < st; ++j)
            acc += dt * cAt[st - 1][j] * kxg[(j * NWG + wg) * (ROWS * DD) + e];
          s_xs[li] = acc;
        }
      }
      __syncthreads();
      const float ti = t + cC[st] * dt;

      // ---- GEMM1: h = xs @ W1, th = tanh(h + b1 + ti*u1); 4 N-tiles/wave ----
      {
        v8f acc[4];
#pragma unroll
        for (int q = 0; q < 4; ++q)
#pragma unroll
          for (int r = 0; r < 8; ++r) acc[q][r] = 0.f;
#pragma unroll 4
        for (int k = 0; k < DD; k += 4) {
          v2f a = *(const v2f*)&s_xs[mrow * XP + k + 2 * hi];
          const int kb = k + 2 * hi;
#pragma unroll
          for (int q = 0; q < 4; ++q) {
            const int n = wave * 64 + q * 16 + mrow;
            v2f b;
            b.x = W1[kb * HH + n];
            b.y = W1[(kb + 1) * HH + n];
            acc[q] = wmma_f32(a, b, acc[q]);
          }
        }
#pragma unroll
        for (int q = 0; q < 4; ++q) {
          const int n = wave * 64 + q * 16 + mrow;
          const float bb = b1[n] + ti * u1[n];
#pragma unroll
          for (int r = 0; r < 8; ++r)
            s_th[(r + 8 * hi) * TP + n] = tanhf(acc[q][r] + bb);
        }
      }
      __syncthreads();

      // ---- GEMM2: dxdt = th @ W2 + b2; one 16x16 tile/wave, K=512 ----
      {
        v8f accs[4];
#pragma unroll
        for (int q = 0; q < 4; ++q)
#pragma unroll
          for (int r = 0; r < 8; ++r) accs[q][r] = 0.f;
        const int n = wave * 16 + mrow;
#pragma unroll 2
        for (int k = 0; k < HH; k += 16) {
#pragma unroll
          for (int c = 0; c < 4; ++c) {   // 4 independent accumulator chains
            const int kk = k + 4 * c;
            v2f a = *(const v2f*)&s_th[mrow * TP + kk + 2 * hi];
            const int kb = kk + 2 * hi;
            v2f b;
            b.x = W2[kb * DD + n];
            b.y = W2[(kb + 1) * DD + n];
            accs[c] = wmma_f32(a, b, accs[c]);
          }
        }
        v8f acc = accs[0] + accs[1] + accs[2] + accs[3];
        const float bb = b2[n];
        const int kbase = (st * NWG + wg) * (ROWS * DD);
#pragma unroll
        for (int r = 0; r < 8; ++r) {
          const int m = r + 8 * hi;
          const float v = acc[r] + bb;
          s_kxi[m * XP + n] = v;
          kxg[kbase + m * DD + n] = v;   // volatile: bypass WGP$ for reuse
        }
        asm volatile("s_wait_storecnt 0" ::: "memory");
      }
      __syncthreads();

      // ---- deterministic reductions: tr = sum (1-th^2)c_k ; dot = <xs,dxdt> ----
      {
        const int m = tid >> 4, c16 = tid & 15;
        float trp = 0.f;
        for (int q = 0; q < HH / 16; ++q) {
          const float v = s_th[m * TP + c16 + 16 * q];
          trp += (1.f - v * v) * s_c[c16 + 16 * q];
        }
        float dotp = 0.f;
        for (int q = 0; q < DD / 16; ++q) {
          const int li = m * XP + c16 + 16 * q;
          dotp += s_xs[li] * s_kxi[li];
        }
        s_redtr[tid] = trp;
        s_reddot[tid] = dotp;
      }
      __syncthreads();
      if (tid < ROWS) {
        float tr = 0.f, dp = 0.f;
        for (int q = 0; q < 16; ++q) {
          tr += s_redtr[tid * 16 + q];
          dp += s_reddot[tid * 16 + q];
        }
        s_kld[st * ROWS + tid] = -tr;       // dlogp/dt
        s_kkl[st * ROWS + tid] = dp - tr;   // dkl/dt
      }
      __syncthreads();
    }

    // ---- 5th-order combine ----
    for (int e = tid; e < ROWS * DD; e += TPB) {
      const int li = (e >> 7) * XP + (e & 127);
      float acc = s_xcur[li];
      for (int j = 0; j < 6; ++j)
        acc += dt * cBW[j] * kxg[(j * NWG + wg) * (ROWS * DD) + e];
      s_xcur[li] = acc;
    }
    if (tid < ROWS) {
      float a1 = 0.f, a2 = 0.f;
      for (int j = 0; j < 6; ++j) {
        a1 += cBW[j] * s_kld[j * ROWS + tid];
        a2 += cBW[j] * s_kkl[j * ROWS + tid];
      }
      s_ld[tid] += dt * a1;
      s_kl[tid] += dt * a2;
    }
    __syncthreads();
  }

  // ---- outputs: z (B*D), log_px (B), kl (B) ----
  for (int e = tid; e < ROWS * DD; e += TPB)
    out[row0 * DD + e] = s_xcur[(e >> 7) * XP + (e & 127)];
  if (tid < ROWS) {
    const int brow = row0 + tid;
    float ss = 0.f;
    for (int i = 0; i < DD; ++i) {
      const float xv = x0[brow * DD + i];
      ss += xv * xv;
    }
    const float LOG2PI = 1.8378770664093453f;
    out[BB * DD + brow] = -0.5f * ss - 0.5f * (float)DD * LOG2PI + s_ld[tid];
    out[BB * DD + BB + brow] = s_kl[tid];
  }
}

extern "C" void kernel_launch(void* const* d_in, const int* in_sizes, int n_in,
                              void* d_out, int out_size, void* d_ws, size_t ws_size,
                              hipStream_t stream) {
  const float* x0 = (const float*)d_in[0];
  const float* W1 = (const float*)d_in[1];
  const float* b1 = (const float*)d_in[2];
  const float* u1 = (const float*)d_in[3];
  const float* W2 = (const float*)d_in[4];
  const float* b2 = (const float*)d_in[5];
  // d_in[6] = n_steps (compile-time NSTEPS=16)
  float* out = (float*)d_out;
  volatile float* kxg = (volatile float*)d_ws;  // 6 * 8 * 16 * 128 floats = 384 KB

  VINeuralODE_54769422958763_kernel<<<dim3(NWG), dim3(TPB), 0, stream>>>(
      x0, W1, b1, u1, W2, b2, out, kxg);
}